// Model_32289564131887
// MI455X (gfx1250) — compile-verified
//
#include <hip/hip_runtime.h>

typedef __attribute__((ext_vector_type(2))) float v2f;
typedef __attribute__((ext_vector_type(8))) float v8f;

// ---------------- graph preprocessing ----------------

__global__ void zero3_kernel(int* __restrict__ a, int* __restrict__ b,
                             int* __restrict__ c, int n) {
  int i = blockIdx.x * blockDim.x + threadIdx.x;
  if (i < n) { a[i] = 0; b[i] = 0; c[i] = 0; }
}

__global__ void count_deg_kernel(const int* __restrict__ src, const int* __restrict__ dst,
                                 int* __restrict__ degout, int* __restrict__ degin, int E) {
  int i = blockIdx.x * blockDim.x + threadIdx.x;
  if (i < E) {
    atomicAdd(&degout[src[i]], 1);
    atomicAdd(&degin[dst[i]], 1);
  }
}

__global__ void inv_sqrt_kernel(const int* __restrict__ dout, const int* __restrict__ din,
                                float* __restrict__ invout, float* __restrict__ invin, int n) {
  int i = blockIdx.x * blockDim.x + threadIdx.x;
  if (i < n) {
    int a = dout[i] < 1 ? 1 : dout[i];
    int b = din[i]  < 1 ? 1 : din[i];
    invout[i] = rsqrtf((float)a);
    invin[i]  = rsqrtf((float)b);
  }
}

__device__ __forceinline__ int wave_incl_scan(int v, int lane) {
#pragma unroll
  for (int off = 1; off < 32; off <<= 1) {
    int t = __shfl_up(v, off, 32);
    if (lane >= off) v += t;
  }
  return v;
}

// Single block of 1024 threads (32 waves), 4096 elements per tile.
__global__ void scan_rowptr_kernel(const int* __restrict__ deg, int* __restrict__ rowptr, int n) {
  __shared__ int wsum[32];
  __shared__ int carry_s;
  const int tid  = threadIdx.x;
  const int lane = tid & 31;
  const int wv   = tid >> 5;
  if (tid == 0) { carry_s = 0; rowptr[0] = 0; }
  __syncthreads();
  for (int base = 0; base < n; base += 4096) {
    int idx = base + tid * 4;
    int e0 = (idx + 0 < n) ? deg[idx + 0] : 0;
    int e1 = (idx + 1 < n) ? deg[idx + 1] : 0;
    int e2 = (idx + 2 < n) ? deg[idx + 2] : 0;
    int e3 = (idx + 3 < n) ? deg[idx + 3] : 0;
    int tsum = e0 + e1 + e2 + e3;
    int incl = wave_incl_scan(tsum, lane);
    if (lane == 31) wsum[wv] = incl;
    __syncthreads();
    if (wv == 0) wsum[lane] = wave_incl_scan(wsum[lane], lane);
    __syncthreads();
    int start = carry_s + (wv ? wsum[wv - 1] : 0) + incl - tsum;
    int p0 = start + e0, p1 = p0 + e1, p2 = p1 + e2, p3 = p2 + e3;
    if (idx + 0 < n) rowptr[idx + 1] = p0;
    if (idx + 1 < n) rowptr[idx + 2] = p1;
    if (idx + 2 < n) rowptr[idx + 3] = p2;
    if (idx + 3 < n) rowptr[idx + 4] = p3;
    __syncthreads();
    if (tid == 0) carry_s += wsum[31];
    __syncthreads();
  }
}

__global__ void fill_csr_kernel(const int* __restrict__ dst, const int* __restrict__ rowptr,
                                int* __restrict__ cursor, int* __restrict__ eid, int E) {
  int i = blockIdx.x * blockDim.x + threadIdx.x;
  if (i < E) {
    int d = dst[i];
    int p = atomicAdd(&cursor[d], 1);
    eid[rowptr[d] + p] = i;
  }
}

// ---------------- dense: Out[M,NCOLS] = (A[M,128] * scale[M]) @ W[128,NCOLS] ----------------
// One wave computes one 16x16 tile via V_WMMA_F32_16X16X4_F32, fully unrolled K=128.
// OOB rows/cols are CLAMPED (not predicated): a D element (m,n) depends only on A row m
// and B column n, so clamped garbage only reaches D elements that are never stored.

template <int NCOLS, int LDO>
__global__ void gemm_norm_wmma_kernel(const float* __restrict__ A,
                                      const float* __restrict__ scale,
                                      const float* __restrict__ W,
                                      float* __restrict__ Out,
                                      int n_rows, int tiles_n, int total_tiles) {
  int wid = (blockIdx.x * blockDim.x + threadIdx.x) >> 5;
  if (wid >= total_tiles) return;          // wave-uniform; EXEC stays all-1s for WMMA
  int lane = threadIdx.x & 31;
  int mt = wid / tiles_n;
  int nt = wid - mt * tiles_n;
  int half = lane >> 4;                    // 0: lanes 0-15, 1: lanes 16-31
  int l16  = lane & 15;

  int row  = mt * 16 + l16;                // A-matrix: M = lane%16
  int crow = row < n_rows ? row : (n_rows - 1);
  int bcol = nt * 16 + l16;                // B-matrix: N = lane%16
  int cbcol = bcol < NCOLS ? bcol : (NCOLS - 1);

  float s = scale[crow];
  const float* __restrict__ aptr = A + crow * 128 + 2 * half;   // lanes 16-31: K+2/K+3
  const float* __restrict__ wptr = W + cbcol + 2 * half * NCOLS;

  v8f acc = {};
#pragma unroll
  for (int k = 0; k < 128; k += 4) {
    v2f a = *reinterpret_cast<const v2f*>(aptr + k);   // global_load_b64, imm offset
    a = a * s;
    v2f b;
    b.x = wptr[k * NCOLS];                              // imm offsets (NCOLS const)
    b.y = wptr[k * NCOLS + NCOLS];
    acc = __builtin_amdgcn_wmma_f32_16x16x4_f32(false, a, false, b,
                                                (short)0, acc, false, false);
  }
  if (bcol < NCOLS) {
    int orow0 = mt * 16 + half * 8;        // D: VGPR v -> M=v (lo half) / v+8 (hi half)
    float* __restrict__ op = Out + orow0 * LDO + bcol;
#pragma unroll
    for (int v = 0; v < 8; ++v) {
      if (orow0 + v < n_rows) op[v * LDO] = acc[v];
    }
  }
}

// ---------------- sparse: pull-aggregation, one wave per node ----------------
// Unconditional vector gathers; lanes covering features >= F read in-bounds padding
// garbage that is discarded at the (guarded) store.

template <int VEC, int F, int LDT>
__global__ void agg_kernel(const float* __restrict__ t,
                           const int* __restrict__ rowptr, const int* __restrict__ eid,
                           const int* __restrict__ src, const float* __restrict__ ew,
                           const float* __restrict__ invin, const float* __restrict__ bias,
                           float* __restrict__ out, int ldo, int n_nodes, int do_relu) {
  int node = (blockIdx.x * blockDim.x + threadIdx.x) >> 5;
  if (node >= n_nodes) return;
  int lane = threadIdx.x & 31;
  int f0 = lane * VEC;
  const float* __restrict__ tb = t + f0;

  float acc0 = 0.0f, acc1 = 0.0f, acc2 = 0.0f, acc3 = 0.0f;
  int beg = rowptr[node];
  int end = rowptr[node + 1];
  for (int j = beg; j < end; ++j) {
    int e = eid[j];
    int sidx = src[e];
    float wv = ew[e];
    const float* ts = tb + sidx * LDT;
    if (VEC == 4) {
      float4 q = *reinterpret_cast<const float4*>(ts);   // global_load_b128
      acc0 += q.x * wv; acc1 += q.y * wv; acc2 += q.z * wv; acc3 += q.w * wv;
    } else {
      float2 q = *reinterpret_cast<const float2*>(ts);   // global_load_b64
      acc0 += q.x * wv; acc1 += q.y * wv;
    }
  }
  float inv = invin[node];
  float* __restrict__ op = out + node * ldo + f0;
  if (f0 + 0 < F) {
    float r = acc0 * inv + bias[f0 + 0];
    if (do_relu) r = fmaxf(r, 0.0f);
    op[0] = r;
  }
  if (f0 + 1 < F) {
    float r = acc1 * inv + bias[f0 + 1];
    if (do_relu) r = fmaxf(r, 0.0f);
    op[1] = r;
  }
  if (VEC == 4) {
    if (f0 + 2 < F) {
      float r = acc2 * inv + bias[f0 + 2];
      if (do_relu) r = fmaxf(r, 0.0f);
      op[2] = r;
    }
    if (f0 + 3 < F) {
      float r = acc3 * inv + bias[f0 + 3];
      if (do_relu) r = fmaxf(r, 0.0f);
      op[3] = r;
    }
  }
}

// ---------------- driver ----------------

extern "C" void kernel_launch(void* const* d_in, const int* in_sizes, int n_in,
                              void* d_out, int out_size, void* d_ws, size_t ws_size,
                              hipStream_t stream) {
  const float* x   = (const float*)d_in[0];
  const int*   src = (const int*)d_in[1];
  const int*   dst = (const int*)d_in[2];
  const float* w   = (const float*)d_in[3];
  const float* W0  = (const float*)d_in[4];
  const float* b0  = (const float*)d_in[5];
  const float* W1  = (const float*)d_in[6];
  const float* b1  = (const float*)d_in[7];
  const float* W2  = (const float*)d_in[8];
  const float* b2  = (const float*)d_in[9];

  const int FIN = 128;
  const int N = in_sizes[0] / FIN;   // 100000
  const int E = in_sizes[1];         // 800000
  const int H = in_sizes[5];         // 128
  const int C = in_sizes[9];         // 47
  float* out = (float*)d_out;

  char* p = (char*)d_ws;
  auto carve = [&](size_t bytes) -> char* {
    char* r = p;
    p += (bytes + 255) & ~(size_t)255;
    return r;
  };
  int*   degout = (int*)  carve((size_t)N * 4);
  int*   degin  = (int*)  carve((size_t)N * 4);
  float* invout = (float*)carve((size_t)N * 4);
  float* invin  = (float*)carve((size_t)N * 4);
  int*   rowptr = (int*)  carve(((size_t)N + 1) * 4);
  int*   cursor = (int*)  carve((size_t)N * 4);
  int*   eid    = (int*)  carve((size_t)E * 4);
  float* bufA   = (float*)carve((size_t)N * H * 4);
  float* bufB   = (float*)carve((size_t)N * H * 4);

  const int T = 256;
  int blocksN = (N + T - 1) / T;
  int blocksE = (E + T - 1) / T;

  zero3_kernel      <<<blocksN, T, 0, stream>>>(degout, degin, cursor, N);
  count_deg_kernel  <<<blocksE, T, 0, stream>>>(src, dst, degout, degin, E);
  inv_sqrt_kernel   <<<blocksN, T, 0, stream>>>(degout, degin, invout, invin, N);
  scan_rowptr_kernel<<<1, 1024, 0, stream>>>(degin, rowptr, N);
  fill_csr_kernel   <<<blocksE, T, 0, stream>>>(dst, rowptr, cursor, eid, E);

  int tiles_m = (N + 15) / 16;
  int aggBlocks = (int)(((size_t)N * 32 + T - 1) / T);

  // layer 0
  {
    int tiles_n = (H + 15) / 16;
    int total = tiles_m * tiles_n;
    int blocks = (int)(((size_t)total * 32 + T - 1) / T);
    gemm_norm_wmma_kernel<128, 128><<<blocks, T, 0, stream>>>(x, invout, W0, bufA, N, tiles_n, total);
    agg_kernel<4, 128, 128><<<aggBlocks, T, 0, stream>>>(bufA, rowptr, eid, src, w,
                                                         invin, b0, bufB, H, N, 1);
  }
  // layer 1
  {
    int tiles_n = (H + 15) / 16;
    int total = tiles_m * tiles_n;
    int blocks = (int)(((size_t)total * 32 + T - 1) / T);
    gemm_norm_wmma_kernel<128, 128><<<blocks, T, 0, stream>>>(bufB, invout, W1, bufA, N, tiles_n, total);
    agg_kernel<4, 128, 128><<<aggBlocks, T, 0, stream>>>(bufA, rowptr, eid, src, w + (size_t)E,
                                                         invin, b1, bufB, H, N, 1);
  }
  // layer 2 (47 output features, padded ld=48 in scratch)
  {
    int tiles_n = (C + 15) / 16;
    int total = tiles_m * tiles_n;
    int blocks = (int)(((size_t)total * 32 + T - 1) / T);
    gemm_norm_wmma_kernel<47, 48><<<blocks, T, 0, stream>>>(bufB, invout, W2, bufA, N, tiles_n, total);
    agg_kernel<2, 47, 48><<<aggBlocks, T, 0, stream>>>(bufA, rowptr, eid, src, w + 2 * (size_t)E,
                                                       invin, b2, out, C, N, 0);
  }
}